// MSTGCNBlock_9311489097890
// MI455X (gfx1250) — compile-verified
//
#include <hip/hip_runtime.h>
#include <hip/hip_bf16.h>
#include <cstdint>

// ---- problem constants (match reference) ----
#define N_NODES 2000
#define N_EDGES 32000
#define BATCH   8
#define TLEN    12
#define CIN     32
#define KCH     3
#define FCH     64          // F_CHEB == F_T
#define TBDIM   96          // TLEN*BATCH
#define MROWS   (N_NODES*TBDIM)   // 192000 rows for both GEMMs
#define K1      (KCH*CIN)         // 96
#define K2      224               // 3*64 temporal taps + 32 residual
#define K2P     226               // LDS row stride: 226 mod 64 banks -> conflict-free
#define LN_EPS  1e-5f

typedef float v2f __attribute__((ext_vector_type(2)));
typedef float v8f __attribute__((ext_vector_type(8)));

// ---------------------------------------------------------------------------
// 1) degree over source rows (float atomics)
__global__ void mstgcn_deg(const int* __restrict__ ei, const float* __restrict__ ew,
                           float* __restrict__ deg) {
  int e = blockIdx.x * blockDim.x + threadIdx.x;
  if (e < N_EDGES) atomicAdd(&deg[ei[e]], ew[e]);
}

// 2) scaled laplacian weights: edges then self-loops (E+N entries)
__global__ void mstgcn_buildw(const float* __restrict__ ew, const float* __restrict__ deg,
                              const float* __restrict__ lmax, float* __restrict__ wl) {
  int i = blockIdx.x * blockDim.x + threadIdx.x;
  if (i >= N_EDGES + N_NODES) return;
  float inv = 2.0f / lmax[0];
  wl[i] = (i < N_EDGES) ? (-ew[i] * inv) : (deg[i - N_EDGES] * inv - 1.0f);
}

// 3) x0[n][tb][c] gather replaying the reference's permute+reshape flat mapping
__global__ void mstgcn_x0(const float* __restrict__ X, float* __restrict__ x0) {
  size_t i = (size_t)blockIdx.x * blockDim.x + threadIdx.x;
  if (i >= (size_t)MROWS * CIN) return;
  int c  = (int)(i & 31);
  size_t m = i >> 5;
  int tb = (int)(m % TBDIM);
  int n  = (int)(m / TBDIM);
  // flat index into transpose(X,(2,0,1,3)) buffer viewed as [N][C][TB]
  size_t f0 = (size_t)n * (CIN * TBDIM) + (size_t)c * TBDIM + tb;
  int c2 = (int)(f0 / (BATCH * N_NODES * TLEN));
  int r  = (int)(f0 % (BATCH * N_NODES * TLEN));
  int b2 = r / (N_NODES * TLEN);
  int r2 = r % (N_NODES * TLEN);
  int n2 = r2 / TLEN, t2 = r2 % TLEN;
  x0[i] = X[(((size_t)b2 * N_NODES + n2) * CIN + c2) * TLEN + t2];
}

// 4) out[i] = -in[i]  (init for T2 = 2*prop(T1) - T0)
__global__ void mstgcn_neg(const float* __restrict__ in, float* __restrict__ out) {
  size_t i = (size_t)blockIdx.x * blockDim.x + threadIdx.x;
  if (i < (size_t)MROWS * CIN) out[i] = -in[i];
}

// 5) sparse propagation: hout[dst] += scale*w[e]*hin[src], feature dim 3072/node
__global__ void mstgcn_prop(const int* __restrict__ ei, const float* __restrict__ wl,
                            const float* __restrict__ hin, float* __restrict__ hout,
                            float scale) {
  int e = blockIdx.x;                     // 0 .. E+N-1
  int s, d;
  if (e < N_EDGES) { s = ei[e]; d = ei[N_EDGES + e]; }
  else             { s = d = e - N_EDGES; }
  float we = wl[e] * scale;
  const float* src = hin  + (size_t)s * (TBDIM * CIN);
  float*       dst = hout + (size_t)d * (TBDIM * CIN);
  for (int f = threadIdx.x; f < TBDIM * CIN; f += blockDim.x)
    atomicAdd(&dst[f], we * src[f]);
}

// 6) pack W2[224][64]: 3 temporal taps then residual; bias2 = time_b + res_b
__global__ void mstgcn_w2(const float* __restrict__ tw, const float* __restrict__ tb,
                          const float* __restrict__ rw, const float* __restrict__ rb,
                          float* __restrict__ W2, float* __restrict__ b2) {
  int idx = blockIdx.x * blockDim.x + threadIdx.x;
  if (idx < K2 * FCH) {
    int k = idx / FCH, fo = idx % FCH;
    float v;
    if (k < 3 * FCH) { int tap = k >> 6, fi = k & 63; v = tw[(fo * FCH + fi) * 3 + tap]; }
    else             { int ci = k - 3 * FCH;          v = rw[fo * CIN + ci]; }
    W2[idx] = v;
  }
  if (idx < FCH) b2[idx] = tb[idx] + rb[idx];
}

// ---------------------------------------------------------------------------
// 7) Chebyshev GEMM: out[M,64] = relu( Tcat[M, 3x32] @ Wc[96,64] + b )
//    Tcat = {T0|T1|T2} contiguous (term stride MROWS*CIN) so A-fragment loads
//    are pure GLOBAL (saddr+imm) instead of FLAT. One wave per 16-row stripe;
//    all 24 A fragments live in registers across the 4 column tiles.
__global__ void __launch_bounds__(32)
mstgcn_cheb_gemm(const float* __restrict__ Tcat, const float* __restrict__ Wc,
                 const float* __restrict__ bc, float* __restrict__ outc) {
  const int m0   = blockIdx.x * 16;
  const int lane = threadIdx.x;
  const int half = lane >> 4;     // K-half for A/B fragments
  const int lr   = lane & 15;     // A row / B-D column

  // warm WGP$ for this block's A stripe (global_prefetch_b8)
  __builtin_prefetch(Tcat + (size_t)(m0 + lr) * CIN, 0, 1);

  // preload all 24 A fragments (K=96, 4 per k-step). Term index ks>>3 is
  // wave-uniform; only the +2*half K offset is per-lane.
  v2f af[24];
  #pragma unroll
  for (int ks = 0; ks < 24; ++ks) {
    const float* p = Tcat + (size_t)(ks >> 3) * ((size_t)MROWS * CIN)
                          + (size_t)(m0 + lr) * CIN + ((ks * 4) & 31) + 2 * half;
    af[ks].x = p[0]; af[ks].y = p[1];
  }

  #pragma unroll
  for (int ct = 0; ct < 4; ++ct) {
    const int f0 = ct * 16;
    v8f acc = {0.f, 0.f, 0.f, 0.f, 0.f, 0.f, 0.f, 0.f};
    #pragma unroll
    for (int ks = 0; ks < 24; ++ks) {
      int k = ks * 4 + 2 * half;
      v2f bf;
      bf.x = Wc[(size_t)k * FCH + f0 + lr];
      bf.y = Wc[(size_t)(k + 1) * FCH + f0 + lr];
      acc = __builtin_amdgcn_wmma_f32_16x16x4_f32(false, af[ks], false, bf,
                                                  (short)0, acc, false, false);
    }
    float bias = bc[f0 + lr];
    #pragma unroll
    for (int i = 0; i < 8; ++i) {
      float v = acc[i] + bias;
      v = v > 0.f ? v : 0.f;
      outc[(size_t)(m0 + i + 8 * half) * FCH + f0 + lr] = v;
    }
  }
}

// ---------------------------------------------------------------------------
// 8) temporal conv (1x3) + residual (1x1) as one WMMA GEMM, fused ReLU+LayerNorm.
//    rows m = (b*N + n)*T + t ; im2col A staged in LDS (padded stride, bank-
//    conflict-free ds_load_b64 fragment reads) with zero-padded taps.
__global__ void __launch_bounds__(32)
mstgcn_stconv_ln(const float* __restrict__ outc, const float* __restrict__ X,
                 const float* __restrict__ W2, const float* __restrict__ b2,
                 const float* __restrict__ lng, const float* __restrict__ lnb,
                 float* __restrict__ out) {
  __shared__ float As[16 * K2P];    // im2col tile 16 x 224 (stride 226)
  __shared__ float Zs[16 * FCH];    // relu(r+h) tile for LayerNorm

  const int m0   = blockIdx.x * 16;
  const int lane = threadIdx.x;
  const int half = lane >> 4;
  const int lr   = lane & 15;

  // cooperative im2col into LDS
  for (int idx = lane; idx < 16 * K2; idx += 32) {
    int r = idx / K2, k = idx % K2;
    int m = m0 + r;
    int b = m / (N_NODES * TLEN);
    int rem = m % (N_NODES * TLEN);
    int n = rem / TLEN, t = rem % TLEN;
    float v;
    if (k < 3 * FCH) {
      int tap = k >> 6, fi = k & 63;
      int tt = t + tap - 1;
      v = (tt >= 0 && tt < TLEN)
            ? outc[((size_t)n * TBDIM + b * TLEN + tt) * FCH + fi] : 0.f;
    } else {
      int ci = k - 3 * FCH;
      v = X[(((size_t)b * N_NODES + n) * CIN + ci) * TLEN + t];
    }
    As[r * K2P + k] = v;
  }
  __syncthreads();

  #pragma unroll
  for (int ct = 0; ct < 4; ++ct) {
    const int f0 = ct * 16;
    v8f acc = {0.f, 0.f, 0.f, 0.f, 0.f, 0.f, 0.f, 0.f};
    for (int ks = 0; ks < K2 / 4; ++ks) {
      int k = ks * 4 + 2 * half;
      v2f a, bf;
      a.x  = As[lr * K2P + k];
      a.y  = As[lr * K2P + k + 1];
      bf.x = W2[(size_t)k * FCH + f0 + lr];
      bf.y = W2[(size_t)(k + 1) * FCH + f0 + lr];
      acc = __builtin_amdgcn_wmma_f32_16x16x4_f32(false, a, false, bf,
                                                  (short)0, acc, false, false);
    }
    float bias = b2[f0 + lr];
    #pragma unroll
    for (int i = 0; i < 8; ++i) {
      float v = acc[i] + bias;
      v = v > 0.f ? v : 0.f;
      Zs[(i + 8 * half) * FCH + f0 + lr] = v;
    }
  }
  __syncthreads();

  // LayerNorm over F=64 per (b,t,n) row; lanes 0..15 each own one row
  if (lane < 16) {
    int m = m0 + lane;
    int b = m / (N_NODES * TLEN);
    int rem = m % (N_NODES * TLEN);
    int n = rem / TLEN, t = rem % TLEN;
    const float* z = &Zs[lane * FCH];
    float mu = 0.f;
    for (int f = 0; f < FCH; ++f) mu += z[f];
    mu *= (1.0f / FCH);
    float var = 0.f;
    for (int f = 0; f < FCH; ++f) { float d = z[f] - mu; var += d * d; }
    var *= (1.0f / FCH);
    float inv = rsqrtf(var + LN_EPS);
    for (int f = 0; f < FCH; ++f) {
      float v = (z[f] - mu) * inv * lng[f] + lnb[f];
      out[(((size_t)b * N_NODES + n) * FCH + f) * TLEN + t] = v;   // [B,N,F,T]
    }
  }
}

// ---------------------------------------------------------------------------
extern "C" void kernel_launch(void* const* d_in, const int* in_sizes, int n_in,
                              void* d_out, int out_size, void* d_ws, size_t ws_size,
                              hipStream_t stream) {
  const float* X    = (const float*)d_in[0];
  const int*   ei   = (const int*)  d_in[1];
  const float* ew   = (const float*)d_in[2];
  const float* lmax = (const float*)d_in[3];
  const float* Wc   = (const float*)d_in[4];   // [3][32][64] == [96][64]
  const float* bc   = (const float*)d_in[5];
  const float* tw   = (const float*)d_in[6];
  const float* tb   = (const float*)d_in[7];
  const float* rw   = (const float*)d_in[8];
  const float* rb   = (const float*)d_in[9];
  const float* lng  = (const float*)d_in[10];
  const float* lnb  = (const float*)d_in[11];
  float* out = (float*)d_out;

  float* ws   = (float*)d_ws;
  float* deg  = ws;                                   // 2048
  float* wl   = deg + 2048;                           // 34048
  float* x0   = wl + 34048;                           // T0  [M,32]  -- Tcat base
  float* t1   = x0 + (size_t)MROWS * CIN;             // T1  (contiguous with T0)
  float* t2   = t1 + (size_t)MROWS * CIN;             // T2  (contiguous with T1)
  float* outc = t2 + (size_t)MROWS * CIN;             // [M,64]
  float* W2   = outc + (size_t)MROWS * FCH;           // [224,64]
  float* b2   = W2 + (size_t)K2 * FCH;                // [64]

  // Laplacian weights
  hipMemsetAsync(deg, 0, N_NODES * sizeof(float), stream);
  mstgcn_deg<<<(N_EDGES + 255) / 256, 256, 0, stream>>>(ei, ew, deg);
  mstgcn_buildw<<<(N_EDGES + N_NODES + 255) / 256, 256, 0, stream>>>(ew, deg, lmax, wl);

  // T0 gather, then Chebyshev recursion via atomic scatter
  mstgcn_x0<<<((size_t)MROWS * CIN + 255) / 256, 256, 0, stream>>>(X, x0);
  hipMemsetAsync(t1, 0, (size_t)MROWS * CIN * sizeof(float), stream);
  mstgcn_prop<<<N_EDGES + N_NODES, 256, 0, stream>>>(ei, wl, x0, t1, 1.0f);
  mstgcn_neg<<<((size_t)MROWS * CIN + 255) / 256, 256, 0, stream>>>(x0, t2);
  mstgcn_prop<<<N_EDGES + N_NODES, 256, 0, stream>>>(ei, wl, t1, t2, 2.0f);

  // packed weights for fused temporal+residual GEMM
  mstgcn_w2<<<(K2 * FCH + 255) / 256, 256, 0, stream>>>(tw, tb, rw, rb, W2, b2);

  // WMMA GEMM 1: Chebyshev mix + bias + relu (Tcat = x0|t1|t2 contiguous)
  mstgcn_cheb_gemm<<<MROWS / 16, 32, 0, stream>>>(x0, Wc, bc, outc);

  // WMMA GEMM 2: temporal conv + residual + relu + LayerNorm, scatter to [B,N,F,T]
  mstgcn_stconv_ln<<<MROWS / 16, 32, 0, stream>>>(outc, X, W2, b2, lng, lnb, out);
}